// GraphVAE_44985487459129
// MI455X (gfx1250) — compile-verified
//
#include <hip/hip_runtime.h>
#include <hip/hip_bf16.h>
#include <math.h>

typedef __attribute__((ext_vector_type(2))) float v2f;
typedef __attribute__((ext_vector_type(8))) float v8f;

// ---------------------------------------------------------------------------
// One 16x16x4 fp32 WMMA step: C += A[16x4] * B[4x16]
// A frag: lanes 0-15 -> K=0,1 ; lanes 16-31 -> K=2,3 (ISA 7.12.2, 32-bit A)
// B frag: mirrored (row of B striped across lanes within a VGPR)
// ---------------------------------------------------------------------------
__device__ __forceinline__ v8f wmma_f32_step(const float* A, int lda, int arow0, int ak0,
                                             const float* B, int ldb, int bk0, int bcol0,
                                             int r, int half, v8f acc) {
  v2f a, b;
  int k0 = 2 * half;
  a.x = A[(arow0 + r) * lda + ak0 + k0];
  a.y = A[(arow0 + r) * lda + ak0 + k0 + 1];
  b.x = B[(bk0 + k0) * ldb + bcol0 + r];
  b.y = B[(bk0 + k0 + 1) * ldb + bcol0 + r];
  return __builtin_amdgcn_wmma_f32_16x16x4_f32(false, a, false, b, (short)0, acc, false, false);
}

// ---------------------------------------------------------------------------
// K1: build A_norm (sym-normalized adjacency), Agt (adj_gt with unit diag), degA
// ---------------------------------------------------------------------------
__global__ void build_graph(const int* __restrict__ ei, const float* __restrict__ adj_gt,
                            float* __restrict__ An, float* __restrict__ Agt,
                            float* __restrict__ degA) {
  __shared__ float A[96 * 96];
  __shared__ float dinv[96];
  int tid = threadIdx.x;
  for (int i = tid; i < 9216; i += 256) A[i] = 0.f;
  __syncthreads();
  for (int e = tid; e < 1024; e += 256) {
    int rr = ei[e], cc = ei[1024 + e];
    A[rr * 96 + cc] = 1.f;            // benign same-value races
  }
  __syncthreads();
  if (tid < 96) A[tid * 96 + tid] = 1.f;   // max(A, I)
  __syncthreads();
  if (tid < 96) {
    float s = 0.f;
    for (int j = 0; j < 96; ++j) s += A[tid * 96 + j];
    dinv[tid] = rsqrtf(s);
  }
  __syncthreads();
  for (int idx = tid; idx < 9216; idx += 256) {
    int i = idx / 96, j = idx % 96;
    An[idx]  = A[idx] * dinv[i] * dinv[j];
    Agt[idx] = (i == j) ? 1.f : adj_gt[idx];
  }
  if (tid < 96) {
    float s = 0.f;
    for (int j = 0; j < 96; ++j) s += (tid == j) ? 1.f : adj_gt[tid * 96 + j];
    degA[tid] = s;
  }
}

// ---------------------------------------------------------------------------
// Generic fp32 WMMA GEMM: one 16x16 tile per wave.  grid*8 waves == tiles.
// ---------------------------------------------------------------------------
__global__ void gemm_tiles(const float* __restrict__ A, int lda,
                           const float* __restrict__ B, int ldb,
                           float* __restrict__ C, int ldc,
                           int ksteps, int ntn) {
  int wave = threadIdx.x >> 5, lane = threadIdx.x & 31;
  int half = lane >> 4, r = lane & 15;
  int tile = blockIdx.x * (blockDim.x >> 5) + wave;
  int tm = tile / ntn, tn = tile % ntn;
  v8f acc = {0.f, 0.f, 0.f, 0.f, 0.f, 0.f, 0.f, 0.f};
  for (int kb = 0; kb < ksteps; ++kb)
    acc = wmma_f32_step(A, lda, tm * 16, kb * 4, B, ldb, kb * 4, tn * 16, r, half, acc);
  int col = tn * 16 + r;
#pragma unroll
  for (int v = 0; v < 8; ++v)
    C[(tm * 16 + v + 8 * half) * ldc + col] = acc[v];
}

// ---------------------------------------------------------------------------
// K3/K5: H = relu(BN(A_norm @ P + bias)); optional column-mean g_vec.
// Single block (8 waves), pre-BN tile kept in LDS (stride 257, conflict-free).
// ---------------------------------------------------------------------------
__global__ void gcn_bn(const float* __restrict__ An, const float* __restrict__ P,
                       const float* __restrict__ bias, const float* __restrict__ gamma,
                       const float* __restrict__ beta, float* __restrict__ Hout,
                       float* __restrict__ gvec) {
  extern __shared__ float Hs[];     // 96 x 257
  const int LDH = 257;
  int tid = threadIdx.x;
  int wave = tid >> 5, lane = tid & 31, half = lane >> 4, r = lane & 15;
  for (int t = wave; t < 96; t += 8) {           // 6x16 tiles, wave-uniform
    int tm = t >> 4, tn = t & 15;
    v8f acc = {0.f, 0.f, 0.f, 0.f, 0.f, 0.f, 0.f, 0.f};
    for (int kb = 0; kb < 24; ++kb)
      acc = wmma_f32_step(An, 96, tm * 16, kb * 4, P, 256, kb * 4, tn * 16, r, half, acc);
    int col = tn * 16 + r;
    float bv = bias[col];
#pragma unroll
    for (int v = 0; v < 8; ++v)
      Hs[(tm * 16 + v + 8 * half) * LDH + col] = acc[v] + bv;
  }
  __syncthreads();
  int j = tid;                                    // 256 columns, one per thread
  float m = 0.f;
  for (int i = 0; i < 96; ++i) m += Hs[i * LDH + j];
  m *= (1.f / 96.f);
  float var = 0.f;
  for (int i = 0; i < 96; ++i) { float d = Hs[i * LDH + j] - m; var += d * d; }
  var *= (1.f / 96.f);
  float sc = rsqrtf(var + 1e-5f) * gamma[j];
  float bt = beta[j];
  float gs = 0.f;
  for (int i = 0; i < 96; ++i) {
    float val = fmaxf((Hs[i * LDH + j] - m) * sc + bt, 0.f);
    Hout[i * 256 + j] = val;
    gs += val;
  }
  if (gvec) gvec[j] = gs * (1.f / 96.f);
}

// ---------------------------------------------------------------------------
// K6: latent head (mu, clipped logvar, z) + first decoder layer hd
// ---------------------------------------------------------------------------
__global__ void latent_dec1(const float* __restrict__ gvec,
                            const float* __restrict__ Wmu, const float* __restrict__ bmu,
                            const float* __restrict__ Wlv, const float* __restrict__ blv,
                            const float* __restrict__ Wd1, const float* __restrict__ bd1,
                            const float* __restrict__ eps, float* __restrict__ hd) {
  __shared__ float zs[64];
  int tid = threadIdx.x;
  if (tid < 64) {
    float mu = bmu[tid], lv = blv[tid];
    for (int j = 0; j < 256; ++j) {
      float g = gvec[j];
      mu += g * Wmu[j * 64 + tid];
      lv += g * Wlv[j * 64 + tid];
    }
    lv = fminf(fmaxf(lv, -4.f), 4.f);
    zs[tid] = mu + eps[tid] * expf(0.5f * lv);
  }
  __syncthreads();
  float acc = bd1[tid];
  for (int k = 0; k < 64; ++k) acc += zs[k] * Wd1[k * 256 + tid];
  hd[tid] = fmaxf(acc, 0.f);
}

// ---------------------------------------------------------------------------
// K7: vec = tanh(hd @ Wd2 + bd2), 4656 logits
// ---------------------------------------------------------------------------
__global__ void dec2(const float* __restrict__ hd, const float* __restrict__ Wd2,
                     const float* __restrict__ bd2, float* __restrict__ vec) {
  int o = blockIdx.x * blockDim.x + threadIdx.x;
  if (o >= 4656) return;
  float acc = bd2[o];
  for (int j = 0; j < 256; ++j) acc += hd[j] * Wd2[j * 4656 + o];
  vec[o] = tanhf(acc);
}

// ---------------------------------------------------------------------------
// K9: persistent MPM. All state LDS-resident (stride 97). Per iter:
//   T[j,a]=max_b B0[a,b]*X[j,b];  edge=Agt0@T (WMMA);  X=(X*nd+edge)/||.||_F
// ---------------------------------------------------------------------------
__global__ void __launch_bounds__(1024) mpm(const float* __restrict__ Agt,
                                            const float* __restrict__ degA,
                                            const float* __restrict__ vec,
                                            float* __restrict__ out) {
  extern __shared__ float smem[];
  const int LD = 97;
  float* Bm   = smem;               // 96*97   B, diag zeroed
  float* ndl  = Bm  + 96 * LD;      // 96*97   node_sim
  float* Xl   = ndl + 96 * LD;      // 96*97
  float* Tl   = Xl  + 96 * LD;      // 96*97
  float* Agm  = Tl  + 96 * LD;      // 96*97   Agt, diag zeroed
  float* part = Agm + 96 * LD;      // 1024
  float* dAs  = part + 1024;        // 96
  float* dBs  = dAs + 96;           // 96
  int tid = threadIdx.x;

  for (int idx = tid; idx < 9216; idx += 1024) {
    int i = idx / 96, j = idx % 96;
    Agm[i * LD + j] = (i == j) ? 0.f : Agt[idx];
    float bm;
    if (i == j) bm = 0.f;
    else {
      int lo = i < j ? i : j, hi = i < j ? j : i;
      int lin = lo * 95 - (lo * (lo - 1)) / 2 + (hi - lo - 1);   // triu index
      bm = 1.f / (1.f + expf(-vec[lin]));
    }
    Bm[i * LD + j] = bm;
    Xl[i * LD + j] = 1.f / 96.f;
  }
  if (tid < 96) dAs[tid] = degA[tid];
  __syncthreads();
  if (tid < 96) {
    float s = 1.f;                                  // diag(B)=1 contributes
    for (int j = 0; j < 96; ++j) s += Bm[tid * LD + j];
    dBs[tid] = s;
  }
  __syncthreads();
  for (int idx = tid; idx < 9216; idx += 1024) {
    int i = idx / 96, j = idx % 96;
    ndl[i * LD + j] = 1.f / (fabsf(dAs[i] - dBs[j]) + 1.f);
  }
  __syncthreads();

  int wave = tid >> 5, lane = tid & 31, half = lane >> 4, r = lane & 15;
  for (int it = 0; it < 50; ++it) {
    // (a) mul-max "matmul": T[j,a] = max_b B0[a,b]*X[j,b]   (X>=0 throughout)
    for (int idx = tid; idx < 9216; idx += 1024) {
      int j = idx / 96, a = idx % 96;
      const float* br = Bm + a * LD;
      const float* xr = Xl + j * LD;
      float mx = 0.f;
      for (int b = 0; b < 96; ++b) mx = fmaxf(mx, br[b] * xr[b]);
      Tl[j * LD + a] = mx;
    }
    __syncthreads();
    // (b) edge = Agt0 @ T via fp32 WMMA; fused in-place X update (tile-exclusive)
    float lsum = 0.f;
    for (int t = wave; t < 36; t += 32) {           // 6x6 tiles, wave-uniform
      int tm = t / 6, tn = t % 6;
      v8f acc = {0.f, 0.f, 0.f, 0.f, 0.f, 0.f, 0.f, 0.f};
      for (int kb = 0; kb < 24; ++kb)
        acc = wmma_f32_step(Agm, LD, tm * 16, kb * 4, Tl, LD, kb * 4, tn * 16, r, half, acc);
      int col = tn * 16 + r;
#pragma unroll
      for (int v = 0; v < 8; ++v) {
        int o = (tm * 16 + v + 8 * half) * LD + col;
        float xn = Xl[o] * ndl[o] + acc[v];
        Xl[o] = xn;
        lsum += xn * xn;
      }
    }
    part[tid] = lsum;
    __syncthreads();
    for (int s = 512; s > 0; s >>= 1) {             // deterministic tree reduce
      if (tid < s) part[tid] += part[tid + s];
      __syncthreads();
    }
    float scale = rsqrtf(part[0]);
    for (int idx = tid; idx < 9216; idx += 1024)
      Xl[(idx / 96) * LD + idx % 96] *= scale;
    __syncthreads();
  }
  for (int idx = tid; idx < 9216; idx += 1024)
    out[idx] = Xl[(idx / 96) * LD + idx % 96];
}

// ---------------------------------------------------------------------------
extern "C" void kernel_launch(void* const* d_in, const int* in_sizes, int n_in,
                              void* d_out, int out_size, void* d_ws, size_t ws_size,
                              hipStream_t stream) {
  (void)in_sizes; (void)n_in; (void)out_size; (void)ws_size;
  const float* x      = (const float*)d_in[0];
  const int*   ei     = (const int*)  d_in[1];
  const float* adj_gt = (const float*)d_in[2];
  const float* W1  = (const float*)d_in[3];
  const float* b1  = (const float*)d_in[4];
  const float* g1  = (const float*)d_in[5];
  const float* be1 = (const float*)d_in[6];
  const float* W2  = (const float*)d_in[7];
  const float* b2  = (const float*)d_in[8];
  const float* g2  = (const float*)d_in[9];
  const float* be2 = (const float*)d_in[10];
  const float* Wmu = (const float*)d_in[11];
  const float* bmu = (const float*)d_in[12];
  const float* Wlv = (const float*)d_in[13];
  const float* blv = (const float*)d_in[14];
  const float* Wd1 = (const float*)d_in[15];
  const float* bd1 = (const float*)d_in[16];
  const float* Wd2 = (const float*)d_in[17];
  const float* bd2 = (const float*)d_in[18];
  const float* eps = (const float*)d_in[19];

  float* ws   = (float*)d_ws;
  float* An   = ws;              // 9216
  float* Agt  = An   + 9216;     // 9216
  float* degA = Agt  + 9216;     // 128 (padded)
  float* P    = degA + 128;      // 24576
  float* H    = P    + 24576;    // 24576
  float* P2   = H    + 24576;    // 24576
  float* H2   = P2   + 24576;    // 24576
  float* gvec = H2   + 24576;    // 256
  float* hd   = gvec + 256;      // 256
  float* vec  = hd   + 256;      // 4656
  float* out  = (float*)d_out;

  build_graph<<<1, 256, 0, stream>>>(ei, adj_gt, An, Agt, degA);
  // P = x @ W1   (96x64 * 64x256): 96 tiles, K=64 -> 16 steps
  gemm_tiles<<<12, 256, 0, stream>>>(x, 64, W1, 256, P, 256, 16, 16);
  gcn_bn<<<1, 256, 96 * 257 * sizeof(float), stream>>>(An, P, b1, g1, be1, H, nullptr);
  // P2 = H @ W2  (96x256 * 256x256): K=256 -> 64 steps
  gemm_tiles<<<12, 256, 0, stream>>>(H, 256, W2, 256, P2, 256, 64, 16);
  gcn_bn<<<1, 256, 96 * 257 * sizeof(float), stream>>>(An, P2, b2, g2, be2, H2, gvec);
  latent_dec1<<<1, 256, 0, stream>>>(gvec, Wmu, bmu, Wlv, blv, Wd1, bd1, eps, hd);
  dec2<<<19, 256, 0, stream>>>(hd, Wd2, bd2, vec);
  size_t mpm_lds = (size_t)(5 * 96 * 97 + 1024 + 96 + 96) * sizeof(float); // ~187 KB < 320 KB
  mpm<<<1, 1024, mpm_lds, stream>>>(Agt, degA, vec, out);
}